// EARLIEST_3246995276264
// MI455X (gfx1250) — compile-verified
//
#include <hip/hip_runtime.h>
#include <hip/hip_bf16.h>
#include <math.h>

typedef __attribute__((ext_vector_type(2))) float v2f;
typedef __attribute__((ext_vector_type(8))) float v8f;

#define T_LEN 4096
#define F_IN  32
#define H_DIM 50
#define G4    200   // 4*H gate rows
#define GP    208   // G4 padded to 13*16
#define NT    13    // N tiles of 16

// ---------------------------------------------------------------------------
// Kernel 1: G[t, 0:200] = X[t, :] @ Wih^T   (pure fp32 WMMA, exact precision)
// One wave computes one 16(M=time) x 16(N=gate) tile, K=32 via 8 chained
// v_wmma_f32_16x16x4_f32. All operands preloaded with uniform control flow
// so the 16 global_load_b64s clause together ahead of the WMMA chain.
// ---------------------------------------------------------------------------
__global__ __launch_bounds__(32)
void precompute_gates_wmma(const float* __restrict__ X,
                           const float* __restrict__ Wih,
                           float* __restrict__ G) {
    const int tm   = blockIdx.x;        // 0..255  (4096/16 time tiles)
    const int tn   = blockIdx.y;        // 0..12   (208/16 gate tiles)
    const int lane = threadIdx.x;       // wave32
    const int half = lane >> 4;         // 0 or 1
    const int l16  = lane & 15;

    const int row_g = tm * 16 + l16;    // X row (always < 4096)
    const int col_g = tn * 16 + l16;    // Wih row / gate index (may be >= 200)

    // Uniform (branch-free) boundary handling: clamp row, scale by 0/1.
    const int   wrow   = (col_g < G4) ? col_g : (G4 - 1);
    const float bscale = (col_g < G4) ? 1.0f : 0.0f;

    const float* __restrict__ xrow = X   + row_g * F_IN + half * 2;
    const float* __restrict__ wr   = Wih + wrow  * F_IN + half * 2;

    // Preload all A and B fragments (8 x b64 each); loads clause + overlap.
    v2f av[8], bv[8];
    #pragma unroll
    for (int i = 0; i < 8; ++i) {
        av[i] = *(const v2f*)(xrow + i * 4);
        v2f b = *(const v2f*)(wr + i * 4);
        b.x *= bscale;
        b.y *= bscale;
        bv[i] = b;
    }

    v8f acc = {};
    #pragma unroll
    for (int i = 0; i < 8; ++i) {
        acc = __builtin_amdgcn_wmma_f32_16x16x4_f32(
            /*neg_a=*/false, av[i], /*neg_b=*/false, bv[i],
            /*c_mod=*/(short)0, acc, /*reuse_a=*/false, /*reuse_b=*/false);
    }

    // C/D layout: vgpr r, lane L -> (M = r + 8*(L/16), N = L%16)
    if (col_g < G4) {
        float* __restrict__ gout = G + (tm * 16 + half * 8) * GP + col_g;
        #pragma unroll
        for (int r = 0; r < 8; ++r)
            gout[r * GP] = acc[r];
    }
}

// ---------------------------------------------------------------------------
// Kernel 2: the serial scan. One persistent 256-thread block (8 wave32).
//   threads   0..199 : one gate row each; Whh row cached in 50 VGPRs
//   threads   0..49  : own c[j] in a register, update h -> LDS
//   thread    200    : controller head (halt)   for the previous step
//   thread    201    : baseline head            for the previous step
// Two barriers per timestep; heads overlap the next step's matvec.
// ---------------------------------------------------------------------------
__device__ __forceinline__ float sigmoidf_(float x) {
    return 1.0f / (1.0f + expf(-x));
}
__device__ __forceinline__ float softplusf_(float x) {   // log(1+exp(x)), stable
    return fmaxf(x, 0.0f) + log1pf(expf(-fabsf(x)));
}

__global__ __launch_bounds__(256)
void earliest_scan(const float* __restrict__ G,
                   const float* __restrict__ Whh,
                   const float* __restrict__ bih,
                   const float* __restrict__ bhh,
                   const float* __restrict__ Wc,  const float* __restrict__ bc,
                   const float* __restrict__ Wb,  const float* __restrict__ bb,
                   const float* __restrict__ Wd,  const float* __restrict__ bd,
                   float* __restrict__ out) {
    __shared__ __align__(16) float h_lds[H_DIM];   // 50 floats, float2-readable
    __shared__ float gates[G4];

    const int j = threadIdx.x;

    // --- per-thread cached weights ---
    float w[H_DIM];                    // Whh row (gate threads)
    if (j < G4) {
        #pragma unroll
        for (int k = 0; k < H_DIM; ++k) w[k] = Whh[j * H_DIM + k];
    }
    const float bsum = (j < G4) ? (bih[j] + bhh[j]) : 0.0f;

    float wcr[H_DIM + 1]; float bcr = 0.0f;   // controller (thread 200)
    if (j == 200) {
        #pragma unroll
        for (int k = 0; k <= H_DIM; ++k) wcr[k] = Wc[k];
        bcr = bc[0];
    }
    float wbr[H_DIM]; float bbr = 0.0f;       // baseline (thread 201)
    if (j == 201) {
        #pragma unroll
        for (int k = 0; k < H_DIM; ++k) wbr[k] = Wb[k];
        bbr = bb[0];
    }

    if (j < H_DIM) h_lds[j] = 0.0f;
    float c_reg = 0.0f;
    __syncthreads();

    float* y_out    = out;                 // [4]
    float* base_out = out + 4;             // [4096]
    float* halt_out = out + 4 + T_LEN;     // [4096]

    const float2* __restrict__ h2 = (const float2*)h_lds;

    for (int t = 0; t < T_LEN; ++t) {
        // h_lds currently holds h_{t-1} (zeros for t==0)
        if (j < G4) {
            float acc = bsum + G[t * GP + j];
            #pragma unroll
            for (int k2 = 0; k2 < H_DIM / 2; ++k2) {   // 25 x ds_load_b64
                const float2 hp = h2[k2];
                acc += w[2 * k2]     * hp.x;
                acc += w[2 * k2 + 1] * hp.y;
            }
            gates[j] = acc;
            if (((j & 15) == 0) && (t + 1 < T_LEN))
                __builtin_prefetch(&G[(t + 1) * GP + j], 0, 1);
        } else if (j == 200 && t > 0) {
            // controller on [h_{t-1}, (t-1)] -> halt[t-1] = -log(sigmoid(z))
            float z = bcr + wcr[H_DIM] * (float)(t - 1);
            #pragma unroll
            for (int k = 0; k < H_DIM; ++k) z += wcr[k] * h_lds[k];
            halt_out[t - 1] = softplusf_(-z);
        } else if (j == 201 && t > 0) {
            float z = bbr;
            #pragma unroll
            for (int k = 0; k < H_DIM; ++k) z += wbr[k] * h_lds[k];
            base_out[t - 1] = z;
        }
        __syncthreads();   // gates ready; heads done reading h_{t-1}

        if (j < H_DIM) {
            const float ig = sigmoidf_(gates[j]);
            const float fg = sigmoidf_(gates[j +     H_DIM]);
            const float gg = tanhf   (gates[j + 2 * H_DIM]);
            const float og = sigmoidf_(gates[j + 3 * H_DIM]);
            c_reg = fg * c_reg + ig * gg;
            h_lds[j] = og * tanhf(c_reg);
        }
        __syncthreads();   // h_lds = h_t
    }

    // Epilogue: heads for t = T-1 and the classifier on h_{T-1}.
    if (j == 200) {
        float z = bcr + wcr[H_DIM] * (float)(T_LEN - 1);
        #pragma unroll
        for (int k = 0; k < H_DIM; ++k) z += wcr[k] * h_lds[k];
        halt_out[T_LEN - 1] = softplusf_(-z);
    } else if (j == 201) {
        float z = bbr;
        #pragma unroll
        for (int k = 0; k < H_DIM; ++k) z += wbr[k] * h_lds[k];
        base_out[T_LEN - 1] = z;
    } else if (j < 4) {
        float y = bd[j];
        #pragma unroll
        for (int k = 0; k < H_DIM; ++k) y += Wd[j * H_DIM + k] * h_lds[k];
        y_out[j] = y;
    }
}

// ---------------------------------------------------------------------------
extern "C" void kernel_launch(void* const* d_in, const int* in_sizes, int n_in,
                              void* d_out, int out_size, void* d_ws, size_t ws_size,
                              hipStream_t stream) {
    const float* X   = (const float*)d_in[0];
    const float* Wih = (const float*)d_in[1];
    const float* Whh = (const float*)d_in[2];
    const float* bih = (const float*)d_in[3];
    const float* bhh = (const float*)d_in[4];
    const float* Wc  = (const float*)d_in[5];
    const float* bc  = (const float*)d_in[6];
    const float* Wb  = (const float*)d_in[7];
    const float* bb  = (const float*)d_in[8];
    const float* Wd  = (const float*)d_in[9];
    const float* bd  = (const float*)d_in[10];

    float* out = (float*)d_out;
    float* G   = (float*)d_ws;     // [4096 x 208] fp32 = 3.25 MB

    dim3 grid1(T_LEN / 16, NT);
    precompute_gates_wmma<<<grid1, 32, 0, stream>>>(X, Wih, G);

    earliest_scan<<<1, 256, 0, stream>>>(G, Whh, bih, bhh,
                                         Wc, bc, Wb, bb, Wd, bd, out);
}